// SNTEMP_flowSim_gw0_24661702214051
// MI455X (gfx1250) — compile-verified
//
#include <hip/hip_runtime.h>
#include <math.h>

// SNTEMP equilibrium temperature + K1/K2 exchange coefficients.
// Elementwise over N fp32 elements; 50 Newton iterations each -> VALU-bound.
// Round-2 changes vs round 1:
//  * 4 independent Newton chains interleaved in ONE loop (4-way ILP/wave,
//    v_rcp_f32 TRANS op co-executes against other lanes' FMAs)
//  * quartic rewritten in t = Te+273.16:  F = A t^4 - C t^2 + b1 t + b0
//    -> 8 ops/iter/lane instead of ~12; K1 = F'(t_end), Hi = F(t0) reuse it.

#define NEWTON_ITERS 50
#define TKOFF  273.16f
#define TKOFF2 74616.3856f   // 273.16^2
#define A_CONST 5.4e-8f

__device__ __forceinline__ float fast_rcp(float x) {
    return __builtin_amdgcn_rcpf(x);   // v_rcp_f32, ~1 ulp; Newton self-corrects
}

// Per-lane loop state: Newton variable t plus polynomial coefficients.
struct Coef { float t, C, b0, b1; };

__device__ __forceinline__ Coef sntemp_setup(
    float T_a, float swrad, float e_a, float E, float elev, float T_g,
    float sfr, float albedo, float shade_total, float cf)
{
    // saturation vapor pressure (v_exp_f32) and pressure
    float e_s = 6.108f * __expf(17.26939f * T_a / (237.3f + T_a));
    float P   = 1013.0f - 0.1055f * elev;
    // masked reciprocal denominator exactly as reference
    float denom  = e_s - e_a;
    float denom1 = (denom >= 0.0f) ? denom : 0.0f;
    float denom2 = denom1 + ((denom1 == 0.0f) ? 0.01f : 0.0f);
    float B_c    = 0.00061f * P / denom2;          // precise div (setup only)
    // (T_a + 273.16)^4 via two squarings
    float Tk  = T_a + TKOFF;
    float Tk2 = Tk * Tk;
    float Tk4 = Tk2 * Tk2;
    float one_m_sh = 1.0f - shade_total;
    float H_a = (3.354939e-8f + 2.74995e-9f * sqrtf(e_a)) * one_m_sh
              * (1.0f + 0.17f * cf * cf) * Tk4;
    float H_s = (1.0f - albedo) * one_m_sh * swrad;
    float H_v = 0.9f * 5.670373e-8f * sfr * Tk4;   // EMISS_VEG * STB
    float E6  = 1e6f * E;
    float Bq  = E6 * (B_c * (2495.0f + 2.36f * T_a) - 2.36f) + 1.65f; // +K_G/dZ
    float C   = E6 * B_c * 2.36f;
    float D   = H_a + H_s + H_v + 2495.0f * E6 * (B_c * T_a - 1.0f) + T_g * 1.65f;

    Coef c;
    c.t  = Tk;                                     // initial guess = T_a (+offset)
    c.C  = C;
    c.b1 = fmaf(2.0f * TKOFF, C, Bq);              // B + 2*C*273.16
    c.b0 = -fmaf(C, TKOFF2, fmaf(Bq, TKOFF, D));   // -(C*273.16^2 + B*273.16 + D)
    return c;
}

// One interleaved Newton step for one lane (on t-form polynomial).
__device__ __forceinline__ void newton_step(Coef& c) {
    float t2 = c.t * c.t;
    float F  = fmaf(t2, fmaf(A_CONST, t2, -c.C), fmaf(c.b1, c.t, c.b0));
    float Fp = fmaf(c.t, fmaf(4.0f * A_CONST, t2, -2.0f * c.C), c.b1);
    c.t = fmaf(-F, fast_rcp(Fp), c.t);
}

struct Out3 { float te, k1, k2; };

__device__ __forceinline__ Out3 sntemp_epilogue(const Coef& c, float T_0) {
    float t   = c.t;
    float t2  = t * t;
    float K1  = fmaf(t, fmaf(4.0f * A_CONST, t2, -2.0f * c.C), c.b1); // F'(t)
    float Te  = t - TKOFF;
    float t0  = T_0 + TKOFF;
    float t02 = t0 * t0;
    float Hi  = fmaf(t02, fmaf(A_CONST, t02, -c.C), fmaf(c.b1, t0, c.b0)); // F(t0)
    float delt = T_0 - Te;
    float den  = delt * delt;
    float den1 = den + ((den <= 1e-10f) ? 1.0f : 0.0f);
    float K2   = (K1 * delt - Hi) / den1;          // precise div (epilogue only)
    K2 = (fabsf(delt) < 1e-10f) ? 0.0f : K2;
    Out3 r; r.te = Te; r.k1 = K1; r.k2 = K2;
    return r;
}

// ---------------- vectorized kernel: 4 interleaved elements / thread --------

#define SETUP(c) sntemp_setup(ta.c, sw.c, ea.c, ee.c, el.c, tg.c, sf.c, al.c, sh.c, cf.c)

__global__ __launch_bounds__(256)
void sntemp_vec4_kernel(const float* __restrict__ T_a, const float* __restrict__ swrad,
                        const float* __restrict__ e_a, const float* __restrict__ E,
                        const float* __restrict__ elev, const float* __restrict__ T_g,
                        const float* __restrict__ sfr, const float* __restrict__ albedo,
                        const float* __restrict__ shade_total, const float* __restrict__ cfrac,
                        const float* __restrict__ T_0, float* __restrict__ out, int N)
{
    const int N4 = N >> 2;
    const int stride = gridDim.x * blockDim.x;
    for (int i = blockIdx.x * blockDim.x + threadIdx.x; i < N4; i += stride) {
        float4 ta = ((const float4*)T_a)[i];
        float4 sw = ((const float4*)swrad)[i];
        float4 ea = ((const float4*)e_a)[i];
        float4 ee = ((const float4*)E)[i];
        float4 el = ((const float4*)elev)[i];
        float4 tg = ((const float4*)T_g)[i];
        float4 sf = ((const float4*)sfr)[i];
        float4 al = ((const float4*)albedo)[i];
        float4 sh = ((const float4*)shade_total)[i];
        float4 cf = ((const float4*)cfrac)[i];
        float4 t0 = ((const float4*)T_0)[i];

        Coef c[4];
        c[0] = SETUP(x);
        c[1] = SETUP(y);
        c[2] = SETUP(z);
        c[3] = SETUP(w);

        // ONE loop, 4 independent dependency chains interleaved.
#pragma unroll 5
        for (int it = 0; it < NEWTON_ITERS; ++it) {
#pragma unroll
            for (int j = 0; j < 4; ++j) newton_step(c[j]);
        }

        Out3 r0 = sntemp_epilogue(c[0], t0.x);
        Out3 r1 = sntemp_epilogue(c[1], t0.y);
        Out3 r2 = sntemp_epilogue(c[2], t0.z);
        Out3 r3 = sntemp_epilogue(c[3], t0.w);

        ((float4*)(out        ))[i] = make_float4(r0.te, r1.te, r2.te, r3.te);
        ((float4*)(out +     N))[i] = make_float4(r0.k1, r1.k1, r2.k1, r3.k1);
        ((float4*)(out + 2 * N))[i] = make_float4(r0.k2, r1.k2, r2.k2, r3.k2);
    }
}

// ---------------- scalar fallback (N % 4 != 0) ------------------------------

__global__ __launch_bounds__(256)
void sntemp_scalar_kernel(const float* __restrict__ T_a, const float* __restrict__ swrad,
                          const float* __restrict__ e_a, const float* __restrict__ E,
                          const float* __restrict__ elev, const float* __restrict__ T_g,
                          const float* __restrict__ sfr, const float* __restrict__ albedo,
                          const float* __restrict__ shade_total, const float* __restrict__ cfrac,
                          const float* __restrict__ T_0, float* __restrict__ out, int N)
{
    const int stride = gridDim.x * blockDim.x;
    for (int i = blockIdx.x * blockDim.x + threadIdx.x; i < N; i += stride) {
        Coef c = sntemp_setup(T_a[i], swrad[i], e_a[i], E[i], elev[i], T_g[i],
                              sfr[i], albedo[i], shade_total[i], cfrac[i]);
#pragma unroll 5
        for (int it = 0; it < NEWTON_ITERS; ++it) newton_step(c);
        Out3 r = sntemp_epilogue(c, T_0[i]);
        out[i]         = r.te;
        out[N + i]     = r.k1;
        out[2 * N + i] = r.k2;
    }
}

// ---------------- launcher ---------------------------------------------------

extern "C" void kernel_launch(void* const* d_in, const int* in_sizes, int n_in,
                              void* d_out, int out_size, void* d_ws, size_t ws_size,
                              hipStream_t stream)
{
    // setup_inputs() dict order:
    // 0:T_a 1:swrad 2:e_a 3:E 4:elev 5:slope(dead) 6:top_width(dead)
    // 7:up_inflow(dead) 8:T_g 9:shade_fraction_riparian 10:albedo
    // 11:shade_total 12:cloud_fraction 13:T_0
    const float* T_a   = (const float*)d_in[0];
    const float* swrad = (const float*)d_in[1];
    const float* e_a   = (const float*)d_in[2];
    const float* E     = (const float*)d_in[3];
    const float* elev  = (const float*)d_in[4];
    const float* T_g   = (const float*)d_in[8];
    const float* sfr   = (const float*)d_in[9];
    const float* alb   = (const float*)d_in[10];
    const float* shd   = (const float*)d_in[11];
    const float* cfr   = (const float*)d_in[12];
    const float* T_0   = (const float*)d_in[13];
    float* out = (float*)d_out;
    const int N = in_sizes[0];

    const int block = 256;  // 8 wave32 waves per block
    if ((N & 3) == 0) {
        int n4 = N >> 2;
        int grid = (n4 + block - 1) / block;
        if (grid < 1) grid = 1;
        sntemp_vec4_kernel<<<grid, block, 0, stream>>>(
            T_a, swrad, e_a, E, elev, T_g, sfr, alb, shd, cfr, T_0, out, N);
    } else {
        int grid = (N + block - 1) / block;
        if (grid < 1) grid = 1;
        sntemp_scalar_kernel<<<grid, block, 0, stream>>>(
            T_a, swrad, e_a, E, elev, T_g, sfr, alb, shd, cfr, T_0, out, N);
    }
}